// Model_20048907337801
// MI455X (gfx1250) — compile-verified
//
#include <hip/hip_runtime.h>
#include <math.h>

#define ND    100
#define D     16
#define FEAT  64
#define NND   256
#define GR    20000
#define TOBS  2016
#define TTOT  2160
#define TFC   144
#define W2C   40000   // 2*GR columns in W2

typedef __attribute__((ext_vector_type(2))) float v2f;
typedef __attribute__((ext_vector_type(8))) float v8f;

// hardware v_rcp_f32 (~1 ulp) instead of IEEE division sequence
__device__ __forceinline__ float fastrcp_(float x) { return __builtin_amdgcn_rcpf(x); }
__device__ __forceinline__ float sigmoidf_(float x) { return fastrcp_(1.0f + __expf(-x)); }

// workspace layout (float element offsets)
#define WS_H    0                              // normalized H: 16 x 20000
#define WS_HALL (WS_H    + D * GR)             // h = sigmoid(feat@W1+b1): 2160 x 256
#define WS_X    (WS_HALL + TTOT * NND)         // X scan states: 2016 x 16
#define WS_HRH  (WS_X    + TOBS * D)           // HRH: 2016 x 16 x 16
#define WS_HY   (WS_HRH  + TOBS * D * D)       // Hy: 2016 x 16
#define WS_ST   (WS_HY   + TOBS * D)           // forecast states: 144 x 16

// ---------------------------------------------------------------------------
// K0: row-normalize obs_matrix -> H  (16 blocks x 256 threads)
// ---------------------------------------------------------------------------
__global__ void k_norm_H(const float* __restrict__ obs, float* __restrict__ ws) {
    const int d = blockIdx.x;
    const int tid = threadIdx.x;
    __shared__ float red[256];
    __shared__ float rn;
    float s = 0.0f;
    for (int q = tid; q < GR; q += 256) {
        float v = obs[d * GR + q];
        s += v * v;
    }
    red[tid] = s;
    __syncthreads();
    for (int w = 128; w > 0; w >>= 1) {
        if (tid < w) red[tid] += red[tid + w];
        __syncthreads();
    }
    if (tid == 0) rn = 1.0f / sqrtf(red[0]);
    __syncthreads();
    float* H = ws + WS_H;
    for (int q = tid; q < GR; q += 256)
        H[d * GR + q] = obs[d * GR + q] * rn;
}

// ---------------------------------------------------------------------------
// K1: h = sigmoid(features @ W1 + b1) for all 2160 rows (2160 blocks x 256)
// ---------------------------------------------------------------------------
__global__ void k_hidden(const float* __restrict__ feat, const float* __restrict__ W1,
                         const float* __restrict__ b1, float* __restrict__ ws) {
    const int t = blockIdx.x;
    const int j = threadIdx.x;
    __shared__ float fr[FEAT];
    if (j < FEAT) fr[j] = feat[t * FEAT + j];
    __syncthreads();
    float a = b1[j];
#pragma unroll 8
    for (int k = 0; k < FEAT; ++k) a += fr[k] * W1[k * NND + j];
    ws[WS_HALL + t * NND + j] = sigmoidf_(a);
}

// ---------------------------------------------------------------------------
// K2: sequential transition scan X[t] (1 block x 32 threads)
// ---------------------------------------------------------------------------
__global__ void k_xscan(const float* __restrict__ F, const float* __restrict__ ts,
                        const float* __restrict__ init_scale, const float* __restrict__ ninit,
                        const float* __restrict__ ntrans, float* __restrict__ ws) {
    const int tid = threadIdx.x;
    __shared__ float Fs[D * D];
    __shared__ float tss[D];
    __shared__ float xs[D];
    __shared__ float nx[D];
    for (int i = tid; i < D * D; i += 32) Fs[i] = F[i];
    if (tid < D) { tss[tid] = ts[tid]; xs[tid] = init_scale[tid] * ninit[tid]; }
    __syncthreads();
    float* X = ws + WS_X;
    for (int t = 0; t < TOBS; ++t) {
        if (tid < D) {
            float a = 0.0f;
#pragma unroll
            for (int i = 0; i < D; ++i) a += xs[i] * Fs[i * D + tid];
            nx[tid] = a + tss[tid] * ntrans[t * D + tid];
        }
        __syncthreads();
        if (tid < D) { xs[tid] = nx[tid]; X[t * D + tid] = nx[tid]; }
        __syncthreads();
    }
}

// ---------------------------------------------------------------------------
// K3: fused heavy kernel, 16 timesteps per workgroup (8 waves x 32), all four
// matrix stages on v_wmma_f32_16x16x4_f32:
//   logits L = h_tile @ W2_scale_tile       (64 WMMA / wave / round)
//   z      = X_tile @ H_tile                (4 WMMA)
//   HRH_t += (H .* iv_t) @ H^T   (2 t's/wave, 64 WMMA)
//   Hy    += ZIV @ H^T                      (4 WMMA)
// q processed in rounds of 128 (157 rounds; tail masked by iv=ziv=0).
// LDS strides padded for bank-conflict-free lane-indexed rows.
// ---------------------------------------------------------------------------
#define HT_S 260   // h tile stride (16 x 256)
#define HL_S 130   // H chunk stride (16 x 128)
#define IV_S 132   // iv/ziv stride  (16 x 128)
#define XS_S 18    // X tile stride  (16 x 16)

__global__ void __launch_bounds__(256) k_heavy(const float* __restrict__ W2,
                                               const float* __restrict__ b2,
                                               const float* __restrict__ ngr,
                                               float* __restrict__ ws) {
    const int tid  = threadIdx.x;
    const int wave = tid >> 5;
    const int lane = tid & 31;
    const int tbase = blockIdx.x * 16;

    __shared__ float hT[16 * HT_S];     // h[t][k], t local
    __shared__ float Xs[16 * XS_S];     // X[t][d]
    __shared__ float Hl[16 * HL_S];     // H[d][qq], qq in round
    __shared__ float ivS[16 * IV_S];    // iv[t][qq]
    __shared__ float zivS[16 * IV_S];   // ziv[t][qq]
    __shared__ float hyRed[8 * 256];    // per-wave Hy partials

    for (int i = tid; i < 16 * NND; i += 256)
        hT[(i >> 8) * HT_S + (i & 255)] = ws[WS_HALL + (tbase + (i >> 8)) * NND + (i & 255)];
    for (int i = tid; i < 256; i += 256)
        Xs[(i >> 4) * XS_S + (i & 15)] = ws[WS_X + (tbase + (i >> 4)) * D + (i & 15)];
    __syncthreads();

    const int trow = lane & 15;               // M row (t or d) for A operands
    const int n    = lane & 15;               // N col for B/C operands
    const int koff = (lane < 16) ? 0 : 2;     // K offset per half-wave
    const int mb   = (lane < 16) ? 0 : 8;     // C/D row base

    v8f hrA = {};   // HRH accumulator for local t = 2*wave
    v8f hrB = {};   // HRH accumulator for local t = 2*wave+1
    v8f hyA = {};   // Hy partial (16t x 16d) for this wave's q-tiles
    const float* Hws = ws + WS_H;

    for (int q0 = 0; q0 < 20096; q0 += 128) {   // 157 rounds covering 20000
        // stage H chunk 16d x 128q
        for (int i = tid; i < 16 * 128; i += 256) {
            int dd = i >> 7, qq = i & 127;
            int qs = q0 + qq; if (qs >= GR) qs = GR - 1;
            Hl[dd * HL_S + qq] = Hws[dd * GR + qs];
        }
        __syncthreads();

        const int qg = q0 + wave * 16 + n;          // this lane's global q
        const bool valid = (qg < GR);
        const int qc = valid ? qg : (GR - 1);       // clamped column

        // ---- 1) scale logits: L = h_tile @ W2[:, GR+q] ----
        v8f lc = {};
#pragma unroll 4
        for (int k0 = 0; k0 < NND; k0 += 4) {
            v2f a, b;
            a.x = hT[trow * HT_S + k0 + koff];
            a.y = hT[trow * HT_S + k0 + koff + 1];
            b.x = W2[(k0 + koff) * W2C + GR + qc];
            b.y = W2[(k0 + koff + 1) * W2C + GR + qc];
            lc = __builtin_amdgcn_wmma_f32_16x16x4_f32(false, a, false, b, (short)0, lc, false, false);
        }
        // ---- z = X_tile @ H_tile ----
        v8f zc = {};
#pragma unroll
        for (int k0 = 0; k0 < D; k0 += 4) {
            v2f a, b;
            a.x = Xs[trow * XS_S + k0 + koff];
            a.y = Xs[trow * XS_S + k0 + koff + 1];
            b.x = Hl[(k0 + koff) * HL_S + wave * 16 + n];
            b.y = Hl[(k0 + koff + 1) * HL_S + wave * 16 + n];
            zc = __builtin_amdgcn_wmma_f32_16x16x4_f32(false, a, false, b, (short)0, zc, false, false);
        }
        // ---- elementwise: scale, iv, ziv (C layout: row=mb+i, col=n) ----
        const float bq = b2[GR + qc];
#pragma unroll
        for (int i = 0; i < 8; ++i) {
            const int tl = mb + i;
            const float scale = 10.0f * sigmoidf_(lc[i] + bq - 2.302585093f);
            float iv = fastrcp_(scale * scale);
            float z = zc[i] + scale * ngr[(tbase + tl) * GR + qc];
            if (!valid) { iv = 0.0f; z = 0.0f; }
            ivS[tl * IV_S + wave * 16 + n] = iv;
            zivS[tl * IV_S + wave * 16 + n] = z * iv;
        }
        __syncthreads();

        // ---- 2) Hy partial over this wave's q-tile: ZIV @ H^T ----
#pragma unroll
        for (int k0 = 0; k0 < D; k0 += 4) {
            v2f a, b;
            a.x = zivS[trow * IV_S + wave * 16 + k0 + koff];
            a.y = zivS[trow * IV_S + wave * 16 + k0 + koff + 1];
            b.x = Hl[n * HL_S + wave * 16 + k0 + koff];
            b.y = Hl[n * HL_S + wave * 16 + k0 + koff + 1];
            hyA = __builtin_amdgcn_wmma_f32_16x16x4_f32(false, a, false, b, (short)0, hyA, false, false);
        }
        // ---- 3) HRH for 2 local t's, all 8 q-subtiles of this round ----
        const int t0 = wave * 2, t1 = wave * 2 + 1;
#pragma unroll 2
        for (int s = 0; s < 8; ++s) {
            const int qr = s * 16;
#pragma unroll
            for (int k0 = 0; k0 < 16; k0 += 4) {
                const int ks = qr + k0 + koff;
                v2f b, a;
                b.x = Hl[n * HL_S + ks];
                b.y = Hl[n * HL_S + ks + 1];
                a.x = b.x * ivS[t0 * IV_S + ks];
                a.y = b.y * ivS[t0 * IV_S + ks + 1];
                hrA = __builtin_amdgcn_wmma_f32_16x16x4_f32(false, a, false, b, (short)0, hrA, false, false);
                a.x = b.x * ivS[t1 * IV_S + ks];
                a.y = b.y * ivS[t1 * IV_S + ks + 1];
                hrB = __builtin_amdgcn_wmma_f32_16x16x4_f32(false, a, false, b, (short)0, hrB, false, false);
            }
        }
        __syncthreads();   // before next round overwrites Hl/ivS/zivS
    }

    // ---- writeback HRH (C layout: row=mb+i, col=n) ----
    {
        float* R0 = ws + WS_HRH + (tbase + wave * 2) * D * D;
        float* R1 = R0 + D * D;
#pragma unroll
        for (int i = 0; i < 8; ++i) {
            R0[(mb + i) * D + n] = hrA[i];
            R1[(mb + i) * D + n] = hrB[i];
        }
    }
    // ---- reduce Hy partials across waves ----
#pragma unroll
    for (int i = 0; i < 8; ++i) hyRed[wave * 256 + (mb + i) * D + n] = hyA[i];
    __syncthreads();
    {
        float s = 0.0f;
#pragma unroll
        for (int w = 0; w < 8; ++w) s += hyRed[w * 256 + tid];
        ws[WS_HY + (tbase + (tid >> 4)) * D + (tid & 15)] = s;
    }
}

// ---------------------------------------------------------------------------
// K4: Kalman scan (2016 sequential steps, 16x16 Gauss-Jordan inversions),
// then symmetrize + Cholesky + forecast scan. 1 block x 256 threads.
// IEEE division kept here: conditioning of the filter matters.
// ---------------------------------------------------------------------------
__device__ void gj_invert16(float* A, float* I, int r, int c, int tid) {
    for (int p = 0; p < D; ++p) {
        __syncthreads();
        const float ip = 1.0f / A[p * D + p];
        __syncthreads();
        if (r == p) { A[tid] *= ip; I[tid] *= ip; }
        __syncthreads();
        const float f  = (r != p) ? A[r * D + p] : 0.0f;
        const float ap = A[p * D + c];
        const float iq = I[p * D + c];
        __syncthreads();
        if (r != p) { A[tid] -= f * ap; I[tid] -= f * iq; }
    }
    __syncthreads();
}

__global__ void k_kalman(const float* __restrict__ F, const float* __restrict__ ts,
                         const float* __restrict__ init_scale,
                         const float* __restrict__ nsf, float* __restrict__ ws) {
    const int tid = threadIdx.x;       // 0..255
    const int r = tid >> 4, c = tid & 15;
    __shared__ float Fs[D * D], Pm[D * D], T1[D * D], A_[D * D], Iv[D * D], Ss[D * D];
    __shared__ float ms[D], mp[D], vs[D], mn[D], Qd[D], tss[D], xs[D], nx[D];

    Fs[tid] = F[tid];
    Pm[tid] = (r == c) ? init_scale[r] * init_scale[r] : 0.0f;
    if (tid < D) { ms[tid] = 0.0f; tss[tid] = ts[tid]; Qd[tid] = ts[tid] * ts[tid]; }
    __syncthreads();

    for (int t = 0; t < TOBS; ++t) {
        if (tid < D) {
            float a = 0.0f;
#pragma unroll
            for (int k = 0; k < D; ++k) a += Fs[k * D + tid] * ms[k];
            mp[tid] = a;
        }
        {
            float a = 0.0f;
#pragma unroll
            for (int k = 0; k < D; ++k) a += Fs[k * D + r] * Pm[k * D + c];
            T1[tid] = a;
        }
        __syncthreads();
        {
            float a = (r == c) ? Qd[r] : 0.0f;
#pragma unroll
            for (int k = 0; k < D; ++k) a += T1[r * D + k] * Fs[k * D + c];
            A_[tid] = a;
            Iv[tid] = (r == c) ? 1.0f : 0.0f;
        }
        __syncthreads();
        gj_invert16(A_, Iv, r, c, tid);          // Iv = S = inv(P_pred)
        Ss[tid] = Iv[tid];
        A_[tid] = Iv[tid] + ws[WS_HRH + t * D * D + tid];
        Iv[tid] = (r == c) ? 1.0f : 0.0f;
        __syncthreads();
        gj_invert16(A_, Iv, r, c, tid);          // Iv = P_new = inv(S + HRH)
        if (tid < D) {
            float a = 0.0f;
#pragma unroll
            for (int k = 0; k < D; ++k) a += Ss[tid * D + k] * mp[k];
            vs[tid] = a + ws[WS_HY + t * D + tid];
        }
        __syncthreads();
        if (tid < D) {
            float a = 0.0f;
#pragma unroll
            for (int k = 0; k < D; ++k) a += Iv[tid * D + k] * vs[k];
            mn[tid] = a;
        }
        Pm[tid] = Iv[tid];
        __syncthreads();
        if (tid < D) ms[tid] = mn[tid];
        __syncthreads();
    }

    // P_T = 0.5(P + P^T) + 1e-6 I ; Cholesky (lower)
    A_[tid] = 0.5f * (Pm[r * D + c] + Pm[c * D + r]) + ((r == c) ? 1e-6f : 0.0f);
    __syncthreads();
    for (int p = 0; p < D; ++p) {
        if (tid == p * D + p) A_[tid] = sqrtf(A_[tid]);
        __syncthreads();
        if (c == p && r > p) A_[tid] /= A_[p * D + p];
        __syncthreads();
        if (r > p && c > p && c <= r) A_[tid] -= A_[r * D + p] * A_[c * D + p];
        __syncthreads();
    }
    float* ST = ws + WS_ST;
    if (tid < D) {
        float a = ms[tid];
        for (int k = 0; k <= tid; ++k) a += A_[tid * D + k] * nsf[k];
        xs[tid] = a;
        ST[tid] = a;
    }
    __syncthreads();
    for (int step = 1; step < TFC; ++step) {
        if (tid < D) {
            float a = 0.0f;
#pragma unroll
            for (int i = 0; i < D; ++i) a += xs[i] * Fs[i * D + tid];
            nx[tid] = a + tss[tid] * nsf[step * D + tid];
        }
        __syncthreads();
        if (tid < D) { xs[tid] = nx[tid]; ST[step * D + tid] = nx[tid]; }
        __syncthreads();
    }
}

// ---------------------------------------------------------------------------
// K5: forecast output. gr_f = states@H + loc_f + scale_f*noise ;
// out = (1 - sigmoid(gate)) * bounded_exp(rate, 10000).  Grid (40,144) x 256.
// ---------------------------------------------------------------------------
__global__ void k_output(const float* __restrict__ W2, const float* __restrict__ b2,
                         const float* __restrict__ ngfc, const float* __restrict__ ws,
                         float* __restrict__ out) {
    const int t = blockIdx.y;
    const int cell = blockIdx.x * 256 + threadIdx.x;   // 0..9999
    __shared__ float h2[NND];
    __shared__ float st[D];
    h2[threadIdx.x] = ws[WS_HALL + (TOBS + t) * NND + threadIdx.x];
    if (threadIdx.x < D) st[threadIdx.x] = ws[WS_ST + t * D + threadIdx.x];
    __syncthreads();
    if (cell >= ND * ND) return;

    const int qg = cell;            // gate column (channel 0)
    const int qr = 10000 + cell;    // rate column (channel 1)
    float a0 = b2[qg];
    float a1 = b2[GR + qg];
    float a2 = b2[qr];
    float a3 = b2[GR + qr];
#pragma unroll 4
    for (int k = 0; k < NND; ++k) {
        const float hk = h2[k];
        const float* wr = W2 + k * W2C;
        a0 += hk * wr[qg];
        a1 += hk * wr[GR + qg];
        a2 += hk * wr[qr];
        a3 += hk * wr[GR + qr];
    }
    const float* H = ws + WS_H;
    float xg = 0.0f, xr = 0.0f;
#pragma unroll
    for (int d = 0; d < D; ++d) {
        xg += st[d] * H[d * GR + qg];
        xr += st[d] * H[d * GR + qr];
    }
    const float scg = 10.0f * sigmoidf_(a1 - 2.302585093f);
    const float scr = 10.0f * sigmoidf_(a3 - 2.302585093f);
    const float grg = xg + a0 + scg * ngfc[t * GR + qg];
    const float grr = xr + a2 + scr * ngfc[t * GR + qr];
    const float gate = sigmoidf_(grg);
    const float rate = 10000.0f * sigmoidf_(grr - 9.210340372f);
    out[t * (ND * ND) + cell] = (1.0f - gate) * rate;
}

// ---------------------------------------------------------------------------
extern "C" void kernel_launch(void* const* d_in, const int* in_sizes, int n_in,
                              void* d_out, int out_size, void* d_ws, size_t ws_size,
                              hipStream_t stream) {
    const float* features    = (const float*)d_in[0];
    // d_in[1] trip_counts (int) unused by reference forward
    const float* W1          = (const float*)d_in[2];
    const float* b1          = (const float*)d_in[3];
    const float* W2          = (const float*)d_in[4];
    const float* b2          = (const float*)d_in[5];
    const float* init_scale  = (const float*)d_in[6];
    const float* trans_mat   = (const float*)d_in[7];
    const float* trans_scale = (const float*)d_in[8];
    const float* obs_matrix  = (const float*)d_in[9];
    const float* noise_init  = (const float*)d_in[10];
    const float* noise_trans = (const float*)d_in[11];
    const float* noise_gr    = (const float*)d_in[12];
    const float* noise_sfc   = (const float*)d_in[13];
    const float* noise_grfc  = (const float*)d_in[14];
    float* out = (float*)d_out;
    float* ws  = (float*)d_ws;

    k_norm_H<<<16, 256, 0, stream>>>(obs_matrix, ws);
    k_hidden<<<TTOT, 256, 0, stream>>>(features, W1, b1, ws);
    k_xscan<<<1, 32, 0, stream>>>(trans_mat, trans_scale, init_scale, noise_init,
                                  noise_trans, ws);
    k_heavy<<<TOBS / 16, 256, 0, stream>>>(W2, b2, noise_gr, ws);
    k_kalman<<<1, 256, 0, stream>>>(trans_mat, trans_scale, init_scale, noise_sfc, ws);
    k_output<<<dim3(40, TFC), 256, 0, stream>>>(W2, b2, noise_grfc, ws, out);
}